// RoIHeads_24962349924529
// MI455X (gfx1250) — compile-verified
//
#include <hip/hip_runtime.h>
#include <math.h>

// ---------------- problem constants (match reference) ----------------
#define BATCH   8
#define NPROP   8192
#define NCLS    91
#define FG      90          // foreground classes (class 0 dropped)
#define IMGW    800.0f
#define IMGH    800.0f
#define STHR    0.05f
#define NMSTHR  0.5f
#define NDET    100
#define XCLIP   4.135166556742356f   // log(1000/16)

#define TPB     1024        // 32 wave32 waves per workgroup

// ---- gfx1250 async-to-LDS path (builtin proven present in round 1) ----
#if defined(__HIP_DEVICE_COMPILE__) && __has_builtin(__builtin_amdgcn_global_load_async_to_lds_b128)
#define USE_ASYNC 1
typedef int v4i_t __attribute__((ext_vector_type(4)));
typedef __attribute__((address_space(1))) v4i_t* gv4i_p;
typedef __attribute__((address_space(3))) v4i_t* lv4i_p;

static __device__ __forceinline__ void async_ld16(const void* g, void* l) {
    __builtin_amdgcn_global_load_async_to_lds_b128(
        (gv4i_p)(void*)g, (lv4i_p)l, 0, 0);
}
template <int N>
static __device__ __forceinline__ void async_wait() {
#if __has_builtin(__builtin_amdgcn_s_wait_asynccnt)
    __builtin_amdgcn_s_wait_asynccnt(N);   // N is a constant expression here
#else
    if (N == 0) asm volatile("s_wait_asynccnt 0" ::: "memory");
    else        asm volatile("s_wait_asynccnt 1" ::: "memory");
#endif
}
#else
#define USE_ASYNC 0
#endif

// ---- gfx1250 TDM path (Tensor Data Mover) ----
#if defined(__HIP_DEVICE_COMPILE__) && __has_builtin(__builtin_amdgcn_tensor_load_to_lds)
#define USE_TDM 1
typedef unsigned int u32x4_t __attribute__((ext_vector_type(4)));
typedef int i32x4_t __attribute__((ext_vector_type(4)));
typedef int i32x8_t __attribute__((ext_vector_type(8)));
typedef __attribute__((address_space(3))) char* las_p;
#else
#define USE_TDM 0
#endif

// Workspace layout:
//   live  : float  [B][FG][NPROP]   (class-major live scores, -1 = dead)
//   boxes : float4 [B][FG][NPROP]   (clipped decoded boxes)
//   cmax  : uint2  [B][FG]          ({score bits, argmax index} per class)

// ------------------------------------------------------------------
// K1: softmax + box decode + clip + validity mask. One thread per row
// (b,n); class-major stores so consecutive threads write adjacent
// elements -> fully coalesced.
// ------------------------------------------------------------------
__global__ __launch_bounds__(256) void decode_kernel(
    const float* __restrict__ logits,
    const float* __restrict__ reg,
    const float* __restrict__ props,
    float*  __restrict__ live,
    float4* __restrict__ boxes)
{
    int r = blockIdx.x * blockDim.x + threadIdx.x;
    if (r >= BATCH * NPROP) return;
    int b = r / NPROP;
    int n = r - b * NPROP;

    const float* lg = logits + (size_t)r * NCLS;
    const float* rg = reg    + (size_t)r * (NCLS * 4);
    float4 p = ((const float4*)props)[r];

    float mx = lg[0];
    #pragma unroll 7
    for (int c = 1; c < NCLS; ++c) mx = fmaxf(mx, lg[c]);
    float ssum = 0.f;
    #pragma unroll 7
    for (int c = 0; c < NCLS; ++c) ssum += __expf(lg[c] - mx);
    float inv = 1.0f / ssum;

    float w  = p.z - p.x, h  = p.w - p.y;
    float cx = p.x + 0.5f * w, cy = p.y + 0.5f * h;

    for (int c = 1; c < NCLS; ++c) {
        __builtin_prefetch(rg + c * 4 + 64, 0, 1);   // global_prefetch_b8
        float score = __expf(lg[c] - mx) * inv;
        float dx = rg[c * 4 + 0] * 0.1f;             // /WX
        float dy = rg[c * 4 + 1] * 0.1f;             // /WY
        float dw = fminf(rg[c * 4 + 2] * 0.2f, XCLIP);
        float dh = fminf(rg[c * 4 + 3] * 0.2f, XCLIP);
        float pcx = dx * w + cx, pcy = dy * h + cy;
        float pw  = __expf(dw) * w, ph = __expf(dh) * h;
        float x1 = fminf(fmaxf(pcx - 0.5f * pw, 0.f), IMGW);
        float y1 = fminf(fmaxf(pcy - 0.5f * ph, 0.f), IMGH);
        float x2 = fminf(fmaxf(pcx + 0.5f * pw, 0.f), IMGW);
        float y2 = fminf(fmaxf(pcy + 0.5f * ph, 0.f), IMGH);
        bool valid = (score > STHR) && ((x2 - x1) >= 1.f) && ((y2 - y1) >= 1.f);
        size_t idx = ((size_t)b * FG + (c - 1)) * NPROP + n;
        live[idx]  = valid ? score : -1.0f;
        boxes[idx] = make_float4(x1, y1, x2, y2);
    }
}

// ------------------------------------------------------------------
// K2: initial per-(image,class) max+argmax, packed {bits(v), n}.
// ------------------------------------------------------------------
__global__ __launch_bounds__(256) void classmax_kernel(
    const float* __restrict__ live, uint2* __restrict__ cmax)
{
    int seg = blockIdx.x;                       // b*FG + c
    const float* L = live + (size_t)seg * NPROP;
    float bv = -2.f; int bn = 0;
    for (int j = threadIdx.x; j < NPROP; j += 256) {
        float v = L[j];
        if (v > bv) { bv = v; bn = j; }
    }
    __shared__ float sv[256];
    __shared__ int   sn[256];
    sv[threadIdx.x] = bv; sn[threadIdx.x] = bn;
    __syncthreads();
    for (int off = 128; off > 0; off >>= 1) {
        if (threadIdx.x < off) {
            float ov = sv[threadIdx.x + off]; int on = sn[threadIdx.x + off];
            if (ov > sv[threadIdx.x] ||
                (ov == sv[threadIdx.x] && on < sn[threadIdx.x])) {
                sv[threadIdx.x] = ov; sn[threadIdx.x] = on;
            }
        }
        __syncthreads();
    }
    if (threadIdx.x == 0)
        cmax[seg] = make_uint2(__float_as_uint(sv[0]), (unsigned)sn[0]);
}

// ------------------------------------------------------------------
// K3: persistent NMS, one workgroup per image.
//  - preamble: TDM (tensor_load_to_lds) DMAs this image's 720B of
//    per-class {max,argmax} state into LDS, s_wait_tensorcnt, barrier.
//  - each of 100 steps: 128-wide winner reduce over 90 classes, then
//    suppress only the winner class's 8192-entry segment with a
//    double-buffered async-to-LDS pipeline (issue k+1, s_wait_asynccnt 1,
//    consume k), recompute that class's max in the same pass.
// ------------------------------------------------------------------
__global__ __launch_bounds__(TPB) void nms_kernel(
    float* __restrict__ live, const float4* __restrict__ boxes,
    const uint2* __restrict__ cmax, float* __restrict__ out)
{
    const int b   = blockIdx.x;
    const int tid = threadIdx.x;
    float*        Lb = live  + (size_t)b * FG * NPROP;
    const float4* Bb = boxes + (size_t)b * FG * NPROP;

    float* oBoxes  = out;                                // [B][NDET][4]
    float* oScores = out + BATCH * NDET * 4;             // [B][NDET]
    float* oLabels = oScores + BATCH * NDET;
    float* oValid  = oLabels + BATCH * NDET;

    __shared__ unsigned scvn[2 * FG];   // interleaved {score bits, index}
    __shared__ float4 tile[2][TPB];     // double-buffered async stage (32 KB)
    __shared__ float  sv[TPB];
    __shared__ int    sn[TPB];
    __shared__ int    sm[128];
    __shared__ float  wv;               // winner broadcast
    __shared__ int    wc, wn;
    __shared__ float4 wbox;

#if USE_TDM
    if (tid == 0) {
        unsigned laddr = (unsigned)(size_t)(las_p)(void*)&scvn[0];
        unsigned long long gaddr =
            (unsigned long long)(size_t)(const void*)(cmax + (size_t)b * FG);
        u32x4_t g0;
        g0[0] = 1u;                                   // count=1, user mode
        g0[1] = laddr;                                // lds_addr
        g0[2] = (unsigned)(gaddr & 0xffffffffu);      // global_addr[31:0]
        g0[3] = (unsigned)((gaddr >> 32) & 0x1ffffffu) | 0x80000000u; // [56:32] + type=2
        i32x8_t g1;
        g1[0] = 0x20000;                              // data_size=4B (code 2)
        g1[1] = (int)(2u * FG) << 16;                 // tensor_dim0 = 180 dwords
        g1[2] = 1 << 16;                              // tensor_dim1 = 1
        g1[3] = (int)(2u * FG) << 16;                 // tile_dim0 = 180
        g1[4] = 0;                                    // tile_dim1/2 unused
        g1[5] = (int)(2u * FG);                       // tensor_dim0_stride
        g1[6] = 0;
        g1[7] = 0;
        i32x4_t z4 = {0, 0, 0, 0};
#if defined(__clang_major__) && (__clang_major__ >= 23)
        i32x8_t z8 = {0, 0, 0, 0, 0, 0, 0, 0};
        __builtin_amdgcn_tensor_load_to_lds(g0, g1, z4, z4, z8, 0);
#else
        __builtin_amdgcn_tensor_load_to_lds(g0, g1, z4, z4, 0);
#endif
        __builtin_amdgcn_s_wait_tensorcnt(0);
    }
#else
    {
        const unsigned* src = (const unsigned*)(cmax + (size_t)b * FG);
        for (int i = tid; i < 2 * FG; i += TPB) scvn[i] = src[i];
    }
#endif
    __syncthreads();

    for (int det = 0; det < NDET; ++det) {
        // ---- 1. cross-class winner (128-wide tree reduce, tie on flat m) ----
        if (tid < 128) {
            float v = -2.f; int n = 0, m = 0x7fffffff;
            if (tid < FG) {
                v = __uint_as_float(scvn[2 * tid]);
                n = (int)scvn[2 * tid + 1];
                m = n * FG + tid;
            }
            sv[tid] = v; sn[tid] = n; sm[tid] = m;
        }
        __syncthreads();
        for (int off = 64; off > 0; off >>= 1) {
            if (tid < off) {
                float ov = sv[tid + off];
                if (ov > sv[tid] || (ov == sv[tid] && sm[tid + off] < sm[tid])) {
                    sv[tid] = ov; sn[tid] = sn[tid + off]; sm[tid] = sm[tid + off];
                }
            }
            __syncthreads();
        }
        if (tid == 0) {
            float v = sv[0];
            int   m = sm[0];
            int   c = m % FG;
            int   n = sn[0];
            int   o = b * NDET + det;
            if (v > 0.f) {
                float4 wb = Bb[(size_t)c * NPROP + n];
                wbox = wb;
                oBoxes[o * 4 + 0] = wb.x; oBoxes[o * 4 + 1] = wb.y;
                oBoxes[o * 4 + 2] = wb.z; oBoxes[o * 4 + 3] = wb.w;
                oScores[o] = v; oLabels[o] = (float)(c + 1); oValid[o] = 1.0f;
            } else {
                oBoxes[o * 4 + 0] = 0.f; oBoxes[o * 4 + 1] = 0.f;
                oBoxes[o * 4 + 2] = 0.f; oBoxes[o * 4 + 3] = 0.f;
                oScores[o] = 0.f; oLabels[o] = 0.f; oValid[o] = 0.f;
            }
            wv = v; wc = c; wn = n;
        }
        __syncthreads();

        if (wv > 0.f) {
            // ---- 2. suppress winner class segment + recompute its max ----
            const float4 wb = wbox;
            const float  wa = (wb.z - wb.x) * (wb.w - wb.y);
            float*        Lc = Lb + (size_t)wc * NPROP;
            const float4* Bc = Bb + (size_t)wc * NPROP;
            float bv = -2.f; int bn = 0;
            int buf = 0;
#if USE_ASYNC
            async_ld16(Bc + tid, &tile[0][tid]);          // prologue: tile 0
#endif
            #pragma unroll
            for (int base = 0; base < NPROP; base += TPB) {
                int j = base + tid;
#if USE_ASYNC
                if (base + TPB < NPROP) {                 // pipeline next tile
                    async_ld16(Bc + (base + TPB) + tid, &tile[buf ^ 1][tid]);
                    async_wait<1>();                      // wait only tile k
                } else {
                    async_wait<0>();
                }
                float  l = Lc[j];
                float4 q = tile[buf][tid];
                buf ^= 1;
#else
                float  l = Lc[j];
                float4 q = Bc[j];
#endif
                if (l > 0.f) {
                    float xx1 = fmaxf(wb.x, q.x), yy1 = fmaxf(wb.y, q.y);
                    float xx2 = fminf(wb.z, q.z), yy2 = fminf(wb.w, q.w);
                    float iw = fmaxf(xx2 - xx1, 0.f), ih = fmaxf(yy2 - yy1, 0.f);
                    float inter = iw * ih;
                    float qa = (q.z - q.x) * (q.w - q.y);
                    float iou = inter / (wa + qa - inter + 1e-9f);
                    if (iou > NMSTHR || j == wn) { Lc[j] = -1.f; l = -1.f; }
                }
                if (l > bv) { bv = l; bn = j; }   // j ascends per thread -> min idx on tie
            }
            sv[tid] = bv; sn[tid] = bn;
            __syncthreads();
            for (int off = TPB / 2; off > 0; off >>= 1) {
                if (tid < off) {
                    float ov = sv[tid + off]; int on = sn[tid + off];
                    if (ov > sv[tid] || (ov == sv[tid] && on < sn[tid])) {
                        sv[tid] = ov; sn[tid] = on;
                    }
                }
                __syncthreads();
            }
            if (tid == 0) {
                scvn[2 * wc]     = __float_as_uint(sv[0]);
                scvn[2 * wc + 1] = (unsigned)sn[0];
            }
        }
        __syncthreads();
    }
}

// ------------------------------------------------------------------
extern "C" void kernel_launch(void* const* d_in, const int* in_sizes, int n_in,
                              void* d_out, int out_size, void* d_ws, size_t ws_size,
                              hipStream_t stream) {
    const float* logits = (const float*)d_in[0];   // [B,N,91]
    const float* reg    = (const float*)d_in[1];   // [B,N,364]
    const float* props  = (const float*)d_in[2];   // [B,N,4]

    const size_t nCand = (size_t)BATCH * FG * NPROP;
    float*  live  = (float*)d_ws;
    float4* boxes = (float4*)((char*)d_ws + sizeof(float) * nCand);
    uint2*  cmax  = (uint2*)((char*)boxes + sizeof(float4) * nCand);
    float*  out   = (float*)d_out;

    decode_kernel<<<(BATCH * NPROP + 255) / 256, 256, 0, stream>>>(
        logits, reg, props, live, boxes);
    classmax_kernel<<<BATCH * FG, 256, 0, stream>>>(live, cmax);
    nms_kernel<<<BATCH, TPB, 0, stream>>>(live, boxes, cmax, out);
}